// SpGAT_38955353374999
// MI455X (gfx1250) — compile-verified
//
#include <hip/hip_runtime.h>
#include <hip/hip_bf16.h>

#define NN 50000
#define NR 500
#define NE 800000
#define DIM 128
#define ALPHA 0.2f
#define EPSV 1e-12f

typedef __attribute__((ext_vector_type(2))) float v2f;
typedef __attribute__((ext_vector_type(8))) float v8f;

// ---------------------------------------------------------------------------
// C[M x 128] = A[M x 128] * B_rows^T, where row j of "B" is B[j*ldb + boff + k]
// One wave (32 threads) per 16x16 output tile; K=128 via 32x v_wmma_f32_16x16x4_f32.
// A 16x4 f32 operand layout: lane l holds M=l&15; VGPR0:K=(l<16?0:2), VGPR1:K+1.
// B 4x16 mirrors with N=l&15. C/D: VGPR v holds M=v+8*(l>>4), N=l&15.
// ---------------------------------------------------------------------------
__global__ void gemm_abt_wmma(const float* __restrict__ A, const float* __restrict__ B,
                              float* __restrict__ C, int M, int ldb, int boff) {
    const int lane  = threadIdx.x & 31;
    const int mhalf = lane >> 4;        // 0 or 1
    const int l15   = lane & 15;
    int mrow = blockIdx.x * 16 + l15;
    if (mrow >= M) mrow = M - 1;        // clamp: keep EXEC all-ones for WMMA
    const int nrow = blockIdx.y * 16 + l15;     // output col == row of B (128 cols, always full)
    const float* __restrict__ arow = A + (size_t)mrow * DIM;
    const float* __restrict__ brow = B + (size_t)nrow * ldb + boff;
    v8f acc = {};
#pragma unroll
    for (int k0 = 0; k0 < DIM; k0 += 4) {
        v2f av = *(const v2f*)(arow + k0 + 2 * mhalf);
        v2f bv = *(const v2f*)(brow + k0 + 2 * mhalf);
        acc = __builtin_amdgcn_wmma_f32_16x16x4_f32(false, av, false, bv,
                                                    (short)0, acc, false, false);
    }
    const int outcol = blockIdx.y * 16 + l15;
#pragma unroll
    for (int v = 0; v < 8; ++v) {
        int orow = blockIdx.x * 16 + v + 8 * mhalf;
        if (orow < M) C[(size_t)orow * DIM + outcol] = acc[v];
    }
}

// ---------------------------------------------------------------------------
// out_rel = relu(rel @ W_r) + rel   (W_r is K-major 128x128, B[k*128 + n])
// ---------------------------------------------------------------------------
__global__ void gemm_ab_rel_wmma(const float* __restrict__ A, const float* __restrict__ B,
                                 float* __restrict__ out, int M) {
    const int lane  = threadIdx.x & 31;
    const int mhalf = lane >> 4;
    const int l15   = lane & 15;
    int mrow = blockIdx.x * 16 + l15;
    if (mrow >= M) mrow = M - 1;
    const int ncol = blockIdx.y * 16 + l15;
    const float* __restrict__ arow = A + (size_t)mrow * DIM;
    v8f acc = {};
#pragma unroll
    for (int k0 = 0; k0 < DIM; k0 += 4) {
        v2f av = *(const v2f*)(arow + k0 + 2 * mhalf);
        v2f bv;
        bv.x = B[(size_t)(k0 + 2 * mhalf) * DIM + ncol];
        bv.y = B[(size_t)(k0 + 2 * mhalf + 1) * DIM + ncol];
        acc = __builtin_amdgcn_wmma_f32_16x16x4_f32(false, av, false, bv,
                                                    (short)0, acc, false, false);
    }
#pragma unroll
    for (int v = 0; v < 8; ++v) {
        int orow = blockIdx.x * 16 + v + 8 * mhalf;
        if (orow < M) {
            float val = acc[v];
            float r   = A[(size_t)orow * DIM + ncol];
            out[(size_t)orow * DIM + ncol] = (val > 0.f ? val : 0.f) + r;
        }
    }
}

// ---------------------------------------------------------------------------
// s[r] = dot(T[r, :], a_2) for r in [0, rows)  (wave per row, shuffle reduce)
// ---------------------------------------------------------------------------
__global__ void rowdot_kernel(const float* __restrict__ T, const float* __restrict__ a2,
                              float* __restrict__ s, int rows) {
    int gwave = (int)((blockIdx.x * blockDim.x + threadIdx.x) >> 5);
    int lane  = threadIdx.x & 31;
    if (gwave >= rows) return;
    const float4* tp = (const float4*)(T + (size_t)gwave * DIM);
    const float4* ap = (const float4*)a2;
    float4 t = tp[lane];
    float4 av = ap[lane];
    float d = t.x * av.x + t.y * av.y + t.z * av.z + t.w * av.w;
#pragma unroll
    for (int off = 16; off > 0; off >>= 1)
        d += __shfl_xor(d, off, 32);
    if (lane == 0) s[gwave] = d;
}

__global__ void zero_kernel(float* __restrict__ p, size_t n) {
    size_t i = (size_t)blockIdx.x * blockDim.x + threadIdx.x;
    if (i < n) p[i] = 0.f;
}

// ---------------------------------------------------------------------------
// One wave per edge: e = exp(-leaky_relu(s_src[row]+s_dst[col]+s_rel[r]))
//   e_rowsum[row] += e ;  h_num[row, :] += e * (T_dst[col, :] + T_rel[r, :])
// (the e*T_src[row] term is factored out as e_rowsum[row]*T_src[row] later)
// ---------------------------------------------------------------------------
__global__ void edge_kernel(const int* __restrict__ edge_list, const int* __restrict__ rel_list,
                            const float* __restrict__ T_dst, const float* __restrict__ T_rel,
                            const float* __restrict__ s_src, const float* __restrict__ s_dst,
                            const float* __restrict__ s_rel,
                            float* __restrict__ e_rowsum, float* __restrict__ h_num) {
    int gwave = (int)((blockIdx.x * blockDim.x + threadIdx.x) >> 5);
    int lane  = threadIdx.x & 31;
    if (gwave >= NE) return;
    int r0 = edge_list[gwave];
    int c0 = edge_list[NE + gwave];
    int rl = rel_list[gwave];
    float p  = s_src[r0] + s_dst[c0] + s_rel[rl];
    float lr = p > 0.f ? p : ALPHA * p;
    float ee = expf(-lr);
    const float4* td = (const float4*)(T_dst + (size_t)c0 * DIM);
    const float4* tr = (const float4*)(T_rel + (size_t)rl * DIM);
    float4 d = td[lane];
    float4 r = tr[lane];
    float* hn = h_num + (size_t)r0 * DIM + lane * 4;
    atomicAdd(hn + 0, ee * (d.x + r.x));
    atomicAdd(hn + 1, ee * (d.y + r.y));
    atomicAdd(hn + 2, ee * (d.z + r.z));
    atomicAdd(hn + 3, ee * (d.w + r.w));
    if (lane == 0) atomicAdd(e_rowsum + r0, ee);
}

// out_ent = relu((e_rowsum*T_src + h_num) / (e_rowsum + eps))   [relu(elu(x)) == relu(x)]
__global__ void finalize_kernel(const float* __restrict__ T_src, const float* __restrict__ h_num,
                                const float* __restrict__ e_rowsum, float* __restrict__ out) {
    size_t idx = (size_t)blockIdx.x * blockDim.x + threadIdx.x;
    if (idx >= (size_t)NN * DIM) return;
    int i = (int)(idx >> 7);
    float es = e_rowsum[i];
    float v  = (es * T_src[idx] + h_num[idx]) / (es + EPSV);
    out[idx] = v > 0.f ? v : 0.f;
}

extern "C" void kernel_launch(void* const* d_in, const int* in_sizes, int n_in,
                              void* d_out, int out_size, void* d_ws, size_t ws_size,
                              hipStream_t stream) {
    const float* ent      = (const float*)d_in[0];
    const float* rel      = (const float*)d_in[1];
    const int*   edges    = (const int*)d_in[2];   // (2, NE) int32
    const int*   rels     = (const int*)d_in[3];   // (NE,)  int32
    const float* a        = (const float*)d_in[4]; // (128, 384)
    const float* a2       = (const float*)d_in[5]; // (128,)
    const float* W_r      = (const float*)d_in[6]; // (128, 128)

    float* out_ent = (float*)d_out;
    float* out_rel = out_ent + (size_t)NN * DIM;

    // workspace layout (floats): T_src | T_dst | T_rel | s_src | s_dst | s_rel | e_rowsum | h_num
    float* ws      = (float*)d_ws;
    float* T_src   = ws;
    float* T_dst   = T_src + (size_t)NN * DIM;
    float* T_rel   = T_dst + (size_t)NN * DIM;
    float* s_src   = T_rel + (size_t)NR * DIM;
    float* s_dst   = s_src + NN;
    float* s_rel   = s_dst + NN;
    float* e_rowsum= s_rel + NR;
    float* h_num   = e_rowsum + NN;

    // zero the accumulation region (e_rowsum + h_num are contiguous)
    size_t nz = (size_t)NN + (size_t)NN * DIM;
    zero_kernel<<<dim3((unsigned)((nz + 255) / 256)), 256, 0, stream>>>(e_rowsum, nz);

    // node-space GEMMs (WMMA f32 16x16x4)
    dim3 gT((NN + 15) / 16, DIM / 16);
    gemm_abt_wmma<<<gT, 32, 0, stream>>>(ent, a, T_src, NN, 3 * DIM, 0);
    gemm_abt_wmma<<<gT, 32, 0, stream>>>(ent, a, T_dst, NN, 3 * DIM, DIM);
    dim3 gR((NR + 15) / 16, DIM / 16);
    gemm_abt_wmma<<<gR, 32, 0, stream>>>(rel, a, T_rel, NR, 3 * DIM, 2 * DIM);
    gemm_ab_rel_wmma<<<gR, 32, 0, stream>>>(rel, W_r, out_rel, NR);

    // per-row scalar dots with a_2 over the contiguous [T_src; T_dst; T_rel] block
    int rows = 2 * NN + NR;
    rowdot_kernel<<<((size_t)rows * 32 + 255) / 256, 256, 0, stream>>>(T_src, a2, s_src, rows);

    // edge scatter phase (L2-resident gathers + f32 atomics)
    edge_kernel<<<(NE + 7) / 8, 256, 0, stream>>>(edges, rels, T_dst, T_rel,
                                                  s_src, s_dst, s_rel, e_rowsum, h_num);

    // finalize out_ent
    size_t nelt = (size_t)NN * DIM;
    finalize_kernel<<<(nelt + 255) / 256, 256, 0, stream>>>(T_src, h_num, e_rowsum, out_ent);
}